// TopK_69776038690884
// MI455X (gfx1250) — compile-verified
//
#include <hip/hip_runtime.h>
#include <stdint.h>

// TopK activation, rows of 16384 fp32, k from device scalar.
// One block per row; row staged in LDS via CDNA5 async global->LDS copies;
// threshold found by 4-pass radix select on |x| bit patterns with
// privatized histograms, parallel suffix scan, and post-pass-0 compaction.

#define W       16384
#define BLK     256
#define NW      (BLK / 32)        // 8 wave32 per block
#define CHUNKS  (W / 4 / BLK)     // 16 float4 chunks per thread
#define HPAD    260               // per-wave histogram stride (bank-staggered)
#define LISTMAX 2048              // compacted candidate list capacity

typedef float float4v __attribute__((ext_vector_type(4)));
typedef int   v4i     __attribute__((vector_size(16)));   // matches builtin param type

#if __has_builtin(__builtin_amdgcn_global_load_async_to_lds_b128)
#define HAVE_ASYNC_B128 1
#else
#define HAVE_ASYNC_B128 0
#endif
#if __has_builtin(__builtin_amdgcn_s_wait_asynccnt)
#define HAVE_WAIT_ASYNC 1
#else
#define HAVE_WAIT_ASYNC 0
#endif

__global__ __launch_bounds__(BLK) void topk_rows_kernel(const float* __restrict__ x,
                                                        const int* __restrict__ kptr,
                                                        float* __restrict__ out)
{
    __shared__ __align__(16) float s_row[W];        // 64 KB row cache
    __shared__ unsigned s_whist[NW * HPAD];         // 8.3 KB privatized histograms
    __shared__ unsigned s_hist[256];                // reduced totals
    __shared__ unsigned s_scanA[256];               // suffix-scan ping
    __shared__ unsigned s_scanB[256];               // suffix-scan pong
    __shared__ unsigned s_list[LISTMAX];            // pass-0 survivors (keys)
    __shared__ unsigned s_prefix;
    __shared__ unsigned s_kremain;
    __shared__ unsigned s_tietotal;
    __shared__ unsigned s_listcnt;

    const int  tid   = threadIdx.x;
    const int  wslot = (tid >> 5) * HPAD;
    const long row   = blockIdx.x;
    const float* gsrc = x   + row * (long)W;
    float*       gdst = out + row * (long)W;

    // ---------- Stage row into LDS (async DMA path on CDNA5) ----------
#if HAVE_ASYNC_B128
    {
        const char* g = (const char*)gsrc + tid * 16;
        char*       l = (char*)&s_row[0] + tid * 16;
#pragma unroll
        for (int i = 0; i < CHUNKS; ++i) {
            __builtin_amdgcn_global_load_async_to_lds_b128(
                (v4i*)(g + i * (BLK * 16)),
                (v4i*)(l + i * (BLK * 16)),
                0, 0);
        }
#if HAVE_WAIT_ASYNC
        __builtin_amdgcn_s_wait_asynccnt(0);
#else
        asm volatile("s_wait_asynccnt 0" ::: "memory");
#endif
    }
#else
    {
#pragma unroll
        for (int i = 0; i < CHUNKS; ++i) {
            int c = i * BLK + tid;
            *(float4v*)&s_row[c * 4] = *(const float4v*)&gsrc[c * 4];
        }
    }
#endif

    if (tid == 0) {
        int k = *kptr;
        if (k < 0) k = 0;
        if (k > W) k = W;
        s_kremain  = (unsigned)k;
        s_prefix   = 0u;
        s_tietotal = 0u;
        s_listcnt  = 0u;
    }

    // Zero private histograms for pass 0, then make row data + init visible.
    {
#pragma unroll
        for (int i = tid; i < NW * HPAD; i += BLK) s_whist[i] = 0u;
    }
    __syncthreads();

    // Reduce private histograms, suffix-scan from the top, pick crossing bin.
    auto select_bin = [&](unsigned prefix, int shift, bool last) {
        const unsigned kr = s_kremain;              // stale-read safe: barrier below
        unsigned tot = 0u;
#pragma unroll
        for (int w = 0; w < NW; ++w) tot += s_whist[w * HPAD + tid];
        s_hist[tid]  = tot;
        s_scanA[tid] = tot;
        __syncthreads();
        unsigned* src = s_scanA;
        unsigned* dst = s_scanB;
#pragma unroll
        for (int st = 1; st < 256; st <<= 1) {
            dst[tid] = src[tid] + ((tid + st < 256) ? src[tid + st] : 0u);
            __syncthreads();
            unsigned* t = src; src = dst; dst = t;
        }
        const unsigned inc   = src[tid];            // sum of bins >= tid
        const unsigned above = (tid < 255) ? src[tid + 1] : 0u;
        if ((inc >= kr && above < kr) || (kr == 0u && tid == 255)) {
            s_prefix  = prefix | ((unsigned)tid << shift);
            s_kremain = kr - above;
            if (last) s_tietotal = s_hist[tid];
        }
        __syncthreads();
    };

    // ---------- Pass 0: top byte, all elements, 4-way combined atomics ----------
    {
#pragma unroll 2
        for (int i = 0; i < CHUNKS; ++i) {
            int c = i * BLK + tid;
            float4v v = *(const float4v*)&s_row[c * 4];
            unsigned bn[4];
#pragma unroll
            for (int e = 0; e < 4; ++e)
                bn[e] = __float_as_uint(fabsf(v[e])) >> 24;
#pragma unroll
            for (int e = 0; e < 4; ++e) {
                bool first = true;
#pragma unroll
                for (int j = 0; j < e; ++j) first = first && (bn[e] != bn[j]);
                if (first) {
                    unsigned cnt = 1u;
#pragma unroll
                    for (int j = e + 1; j < 4; ++j) cnt += (bn[e] == bn[j]) ? 1u : 0u;
                    atomicAdd(&s_whist[wslot + bn[e]], cnt);
                }
            }
        }
        __syncthreads();
        select_bin(0u, 24, false);
    }

    // ---------- Compact pass-0 survivors (keys with selected top byte) ----------
    const unsigned b0 = s_prefix >> 24;
    {
        unsigned m = 0u;
#pragma unroll 2
        for (int i = 0; i < CHUNKS; ++i) {
            int c = i * BLK + tid;
            float4v v = *(const float4v*)&s_row[c * 4];
#pragma unroll
            for (int e = 0; e < 4; ++e)
                m += ((__float_as_uint(fabsf(v[e])) >> 24) == b0) ? 1u : 0u;
        }
        unsigned base = atomicAdd(&s_listcnt, m);
        unsigned idx = base;
#pragma unroll 2
        for (int i = 0; i < CHUNKS; ++i) {
            int c = i * BLK + tid;
            float4v v = *(const float4v*)&s_row[c * 4];
#pragma unroll
            for (int e = 0; e < 4; ++e) {
                unsigned key = __float_as_uint(fabsf(v[e]));
                if ((key >> 24) == b0) {
                    if (idx < LISTMAX) s_list[idx] = key;
                    ++idx;
                }
            }
        }
    }
    __syncthreads();
    const unsigned nlist    = s_listcnt;
    const bool     use_list = (nlist <= LISTMAX);   // overflow -> full-scan fallback

    // ---------- Passes 1..3 ----------
#pragma unroll 1
    for (int pass = 1; pass < 4; ++pass) {
        const int shift = 24 - 8 * pass;
#pragma unroll
        for (int i = tid; i < NW * HPAD; i += BLK) s_whist[i] = 0u;
        __syncthreads();

        const unsigned prefix = s_prefix;
        const unsigned himask = 0xFFFFFFFFu << (shift + 8);

        if (use_list) {
            for (unsigned i = tid; i < nlist; i += BLK) {
                unsigned key = s_list[i];
                if (((key ^ prefix) & himask) == 0u)
                    atomicAdd(&s_whist[wslot + ((key >> shift) & 0xFFu)], 1u);
            }
        } else {
#pragma unroll 2
            for (int i = 0; i < CHUNKS; ++i) {
                int c = i * BLK + tid;
                float4v v = *(const float4v*)&s_row[c * 4];
#pragma unroll
                for (int e = 0; e < 4; ++e) {
                    unsigned key = __float_as_uint(fabsf(v[e]));
                    if (((key ^ prefix) & himask) == 0u)
                        atomicAdd(&s_whist[wslot + ((key >> shift) & 0xFFu)], 1u);
                }
            }
        }
        __syncthreads();
        select_bin(prefix, shift, pass == 3);
    }

    const unsigned thr   = s_prefix;
    const unsigned kties = s_kremain;

    // ---------- Rare deterministic fixup: more ties than slots ----------
    if (tid == 0 && s_tietotal > kties) {
        unsigned kept = 0u;
        for (int i = 0; i < W; ++i) {
            unsigned key = __float_as_uint(fabsf(s_row[i]));
            if (key == thr) {
                if (kept < kties) ++kept;
                else s_row[i] = 0.0f;               // drop tie; its output is 0 either way
            }
        }
    }
    __syncthreads();

    // ---------- Masked write-out (b128 stores) ----------
#pragma unroll 4
    for (int i = 0; i < CHUNKS; ++i) {
        int c = i * BLK + tid;
        float4v v = *(const float4v*)&s_row[c * 4];
        float4v o;
#pragma unroll
        for (int e = 0; e < 4; ++e) {
            unsigned key = __float_as_uint(fabsf(v[e]));
            o[e] = (key >= thr) ? v[e] : 0.0f;
        }
        *(float4v*)&gdst[c * 4] = o;
    }
}

extern "C" void kernel_launch(void* const* d_in, const int* in_sizes, int n_in,
                              void* d_out, int out_size, void* d_ws, size_t ws_size,
                              hipStream_t stream) {
    const float* x = (const float*)d_in[0];
    const int*   k = (const int*)d_in[1];
    float*     out = (float*)d_out;
    const int rows = in_sizes[0] / W;               // 4096
    topk_rows_kernel<<<rows, BLK, 0, stream>>>(x, k, out);
}